// LmLSTMSharedEmb_23012434772529
// MI455X (gfx1250) — compile-verified
//
#include <hip/hip_runtime.h>
#include <math.h>
#include <stdint.h>

typedef __attribute__((ext_vector_type(2))) float v2f;
typedef __attribute__((ext_vector_type(8))) float v8f;

#define V_   32000
#define E_   512
#define H_   1024
#define G4_  4096   // 4*H
#define B_   16
#define S_   256
#define BS_  4096   // B*S

static __device__ __forceinline__ v8f wmma_f32(v2f a, v2f b, v8f c) {
  // V_WMMA_F32_16X16X4_F32 : D = A(16x4) * B(4x16) + C(16x16)
  return __builtin_amdgcn_wmma_f32_16x16x4_f32(
      /*neg_a=*/false, a, /*neg_b=*/false, b,
      /*c_mod=*/(short)0, c, /*reuse_a=*/false, /*reuse_b=*/false);
}

// -- CDNA5 async global->LDS copy (16B per lane), tracked by ASYNCcnt --------
static __device__ __forceinline__ void async_b128(const float* g, float* lds) {
  // LDS operand is the raw 32-bit LDS offset (generic addr truncated: ISA
  // uses the low 32 bits of the address).
  unsigned int lo = (unsigned int)(uintptr_t)lds;
  unsigned long long ga = (unsigned long long)(uintptr_t)g;
  asm volatile("global_load_async_to_lds_b128 %0, %1, off"
               :: "v"(lo), "v"(ga) : "memory");
}

template <int N>
static __device__ __forceinline__ void wait_async() {
#if defined(__has_builtin)
#if __has_builtin(__builtin_amdgcn_s_wait_asynccnt)
  __builtin_amdgcn_s_wait_asynccnt(N);
#else
  asm volatile("s_wait_asynccnt %0" :: "i"(N) : "memory");
#endif
#else
  asm volatile("s_wait_asynccnt %0" :: "i"(N) : "memory");
#endif
}

// ---------------------------------------------------------------------------
// Init: zero grid barriers, broadcast h0 into ping-pong h-state buffers.
// ---------------------------------------------------------------------------
__global__ __launch_bounds__(256) void init_state(
    const float* __restrict__ h00, const float* __restrict__ h01,
    float* __restrict__ hb0, float* __restrict__ hb1,
    unsigned int* __restrict__ bars) {
  int t = blockIdx.x * blockDim.x + threadIdx.x;
  if (t < 2) bars[t] = 0u;
  for (int i = t; i < B_ * H_; i += gridDim.x * blockDim.x) {
    hb0[i] = h00[i & (H_ - 1)];
    hb1[i] = h01[i & (H_ - 1)];
  }
}

// ---------------------------------------------------------------------------
// Embedding gather: x0[m, :] = emb[tokens[m], :]
// ---------------------------------------------------------------------------
__global__ __launch_bounds__(256) void embed_kernel(
    const int* __restrict__ tok, const float* __restrict__ emb,
    float* __restrict__ x0) {
  int m = blockIdx.x;
  int tk = tok[m];
  const float* s = emb + (size_t)tk * E_;
  float* d = x0 + (size_t)m * E_;
  for (int i = threadIdx.x; i < E_; i += 256) d[i] = s[i];
}

// ---------------------------------------------------------------------------
// fp32 WMMA GEMM: C(MxN) = A(MxK) * B + bias.  BT=0: B is (K,N) row-major,
// fully async-staged + double-buffered.  BT=1: B is (N,K) row-major (logits =
// xproj @ emb^T): A panel async-staged, B transposed through VGPRs.
// Block: 256 threads = 8 waves; block tile 64x64; K-panel 32.
// ---------------------------------------------------------------------------
__global__ __launch_bounds__(256) void gemm_wmma_f32(
    const float* __restrict__ A, const float* __restrict__ Bm,
    const float* __restrict__ bias, float* __restrict__ C,
    int M, int N, int K, int BT) {
  __shared__ float As[2][64][68];   // row stride 272B (16B aligned), pad 4
  __shared__ float Bs[2][32][80];   // row stride 320B (16B aligned), pad 16

  const int tid  = threadIdx.x;
  const int lane = tid & 31;
  const int wave = tid >> 5;
  const int wm   = (wave >> 1) * 16;
  const int wn   = (wave & 1) * 32;
  const int j    = lane & 15;
  const int sel  = (lane >> 4) << 1;   // 0 for lanes 0-15, 2 for lanes 16-31
  const int rb   = (lane >> 4) * 8;    // C-row offset for upper lane half
  const int m0   = blockIdx.y * 64;
  const int n0   = blockIdx.x * 64;

  // staging coordinates
  const int ra = tid >> 3;          // A: rows 0..31 (+32), cols (tid&7)*4
  const int ca = (tid & 7) * 4;
  const int rbv = tid >> 4;         // B: rows 0..15 (+16), cols (tid&15)*4
  const int cbv = (tid & 15) * 4;

  v8f acc0 = {0.f, 0.f, 0.f, 0.f, 0.f, 0.f, 0.f, 0.f};
  v8f acc1 = {0.f, 0.f, 0.f, 0.f, 0.f, 0.f, 0.f, 0.f};

#define GEMM_ISSUE_A(buf, k0)                                                  \
  do {                                                                         \
    async_b128(A + (size_t)(m0 + ra) * K + (k0) + ca, &As[buf][ra][ca]);       \
    async_b128(A + (size_t)(m0 + ra + 32) * K + (k0) + ca,                     \
               &As[buf][ra + 32][ca]);                                         \
  } while (0)

#define GEMM_ISSUE_B(buf, k0)                                                  \
  do {                                                                         \
    async_b128(Bm + (size_t)((k0) + rbv) * N + n0 + cbv, &Bs[buf][rbv][cbv]);  \
    async_b128(Bm + (size_t)((k0) + rbv + 16) * N + n0 + cbv,                  \
               &Bs[buf][rbv + 16][cbv]);                                       \
  } while (0)

#define GEMM_COMPUTE(buf)                                                      \
  do {                                                                         \
    _Pragma("unroll") for (int kk = 0; kk < 32; kk += 4) {                     \
      v2f a;                                                                   \
      a.x = As[buf][wm + j][kk + sel];                                         \
      a.y = As[buf][wm + j][kk + sel + 1];                                     \
      v2f b0, b1;                                                              \
      b0.x = Bs[buf][kk + sel][wn + j];                                        \
      b0.y = Bs[buf][kk + sel + 1][wn + j];                                    \
      b1.x = Bs[buf][kk + sel][wn + 16 + j];                                   \
      b1.y = Bs[buf][kk + sel + 1][wn + 16 + j];                               \
      acc0 = wmma_f32(a, b0, acc0);                                            \
      acc1 = wmma_f32(a, b1, acc1);                                            \
    }                                                                          \
  } while (0)

  if (!BT) {
    // Software-pipelined: panel p+1 streams memory->LDS (ASYNCcnt) while
    // panel p feeds the WMMAs.  4 async ops/thread/panel, so waiting for
    // ASYNCcnt<=4 guarantees the current panel landed.
    const int nPanels = K >> 5;
    GEMM_ISSUE_A(0, 0);
    GEMM_ISSUE_B(0, 0);
    for (int p = 0; p < nPanels; ++p) {
      const int cur = p & 1;
      if (p + 1 < nPanels) {
        const int k0n = (p + 1) << 5;
        GEMM_ISSUE_A(cur ^ 1, k0n);
        GEMM_ISSUE_B(cur ^ 1, k0n);
        wait_async<4>();
      } else {
        wait_async<0>();
      }
      __syncthreads();
      GEMM_COMPUTE(cur);
      __syncthreads();
    }
  } else {
    const int n  = tid & 63;
    const int kb = (tid >> 6) * 8;
    for (int k0 = 0; k0 < K; k0 += 32) {
      GEMM_ISSUE_A(0, k0);
      float4 vb0 = *(const float4*)(Bm + (size_t)(n0 + n) * K + k0 + kb);
      float4 vb1 = *(const float4*)(Bm + (size_t)(n0 + n) * K + k0 + kb + 4);
      Bs[0][kb + 0][n] = vb0.x; Bs[0][kb + 1][n] = vb0.y;
      Bs[0][kb + 2][n] = vb0.z; Bs[0][kb + 3][n] = vb0.w;
      Bs[0][kb + 4][n] = vb1.x; Bs[0][kb + 5][n] = vb1.y;
      Bs[0][kb + 6][n] = vb1.z; Bs[0][kb + 7][n] = vb1.w;
      wait_async<0>();
      __syncthreads();
      GEMM_COMPUTE(0);
      __syncthreads();
    }
  }

  float bv0 = bias ? bias[n0 + wn + j] : 0.f;
  float bv1 = bias ? bias[n0 + wn + 16 + j] : 0.f;
#pragma unroll
  for (int r = 0; r < 8; ++r) {
    int row = m0 + wm + rb + r;
    C[(size_t)row * N + n0 + wn + j]      = acc0[r] + bv0;
    C[(size_t)row * N + n0 + wn + 16 + j] = acc1[r] + bv1;
  }
}

// ---------------------------------------------------------------------------
// Persistent LSTM recurrence. Grid = 64 WGs x 128 threads (4 waves).
// WG owns 16 H-columns; wave g computes gate g's 16x16 z-tile (M = batch = 16)
// via a K=1024 chain of V_WMMA_F32_16X16X4_F32. Device-wide spin barrier
// between timesteps; h state ping-pongs in global (L2-hot); c / prev_out live
// in registers (2 elements per thread).
// ---------------------------------------------------------------------------
__global__ __launch_bounds__(128) void lstm_seq(
    const float* __restrict__ xz,    // (BS, 4H), bias already folded in
    const float* __restrict__ Um,    // (H, 4H)
    const float* __restrict__ c0,    // (H)
    const int*  __restrict__ tok,    // (B, S)
    float* __restrict__ hping,       // 2 * (B*H); buffer 0 pre-initialized
    float* __restrict__ out,         // (BS, H)
    unsigned int* __restrict__ bar) {
  const int wg   = blockIdx.x;      // 0..63
  const int col0 = wg * 16;
  const int tid  = threadIdx.x;
  const int lane = tid & 31;
  const int gate = tid >> 5;        // 0=i 1=f 2=g 3=o
  const int j    = lane & 15;
  const int sel  = (lane >> 4) << 1;
  const int rb   = (lane >> 4) * 8;
  const int nb   = gate * H_ + col0;

  __shared__ float zsh[4][16][16];

  float cs[2], po[2];
#pragma unroll
  for (int u = 0; u < 2; ++u) {
    int e = tid + u * 128;
    cs[u] = c0[col0 + (e & 15)];
    po[u] = 0.f;
  }

  for (int t = 0; t < S_; ++t) {
    const float* hcur  = hping + (size_t)(t & 1) * (B_ * H_);
    float*       hnext = hping + (size_t)((t + 1) & 1) * (B_ * H_);

    // z-tile accumulator initialized with precomputed xz (includes bias)
    v8f acc;
#pragma unroll
    for (int r = 0; r < 8; ++r)
      acc[r] = xz[(size_t)((r + rb) * S_ + t) * G4_ + nb + j];

    const float* arow = hcur + (size_t)j * H_;   // A row = batch index
    for (int k = 0; k < H_; k += 4) {
      v2f a = *(const v2f*)(arow + k + sel);
      v2f b;
      b.x = Um[(size_t)(k + sel) * G4_ + nb + j];
      b.y = Um[(size_t)(k + sel + 1) * G4_ + nb + j];
      acc = wmma_f32(a, b, acc);
    }

#pragma unroll
    for (int r = 0; r < 8; ++r) zsh[gate][rb + r][j] = acc[r];
    __syncthreads();

    // gate combine: 128 threads x 2 elements = 16 rows x 16 cols
#pragma unroll
    for (int u = 0; u < 2; ++u) {
      int e = tid + u * 128;
      int row = e >> 4, cc = e & 15;
      float zi = zsh[0][row][cc];
      float zf = zsh[1][row][cc];
      float zg = zsh[2][row][cc];
      float zo = zsh[3][row][cc];
      float ii = 1.f / (1.f + expf(-zi));
      float ff = 1.f / (1.f + expf(-zf));
      float gg = tanhf(zg);
      float oo = 1.f / (1.f + expf(-zo));
      float cn = ff * cs[u] + ii * gg;
      float hn = oo * tanhf(cn);
      bool  m  = tok[row * S_ + t] != 0;
      float hold = hcur[(size_t)row * H_ + col0 + cc];
      float h2 = m ? hn : hold;
      cs[u]    = m ? cn : cs[u];
      float ov = m ? hn : po[u];
      po[u]    = ov;
      hnext[(size_t)row * H_ + col0 + cc]          = h2;
      out[(size_t)(row * S_ + t) * H_ + col0 + cc] = ov;
    }
    __syncthreads();

    // device-wide barrier before the next step reads hnext
    if (tid == 0) {
      __threadfence();
      __hip_atomic_fetch_add(bar, 1u, __ATOMIC_RELEASE, __HIP_MEMORY_SCOPE_AGENT);
      unsigned int target = (unsigned int)(t + 1) * 64u;
      while (__hip_atomic_load(bar, __ATOMIC_ACQUIRE, __HIP_MEMORY_SCOPE_AGENT) < target)
        __builtin_amdgcn_s_sleep(1);
      __threadfence();
    }
    __syncthreads();
  }
}

// ---------------------------------------------------------------------------
extern "C" void kernel_launch(void* const* d_in, const int* in_sizes, int n_in,
                              void* d_out, int out_size, void* d_ws, size_t ws_size,
                              hipStream_t stream) {
  const int*   tokens = (const int*)d_in[0];
  const float* emb    = (const float*)d_in[1];
  const float* W0     = (const float*)d_in[2];
  const float* U0     = (const float*)d_in[3];
  const float* b0     = (const float*)d_in[4];
  const float* h00    = (const float*)d_in[5];
  const float* c00    = (const float*)d_in[6];
  const float* W1     = (const float*)d_in[7];
  const float* U1     = (const float*)d_in[8];
  const float* b1     = (const float*)d_in[9];
  const float* h01    = (const float*)d_in[10];
  const float* c01    = (const float*)d_in[11];
  const float* pw     = (const float*)d_in[12];
  const float* pb     = (const float*)d_in[13];
  float* logits = (float*)d_out;

  char* ws = (char*)d_ws;
  float* x0    = (float*)(ws);                          // 8 MB  (BS x E)
  float* xzb   = (float*)(ws + (8u  << 20));            // 64 MB (BS x 4H), reused
  float* out0  = (float*)(ws + (72u << 20));            // 16 MB (BS x H)
  float* out1  = (float*)(ws + (88u << 20));            // 16 MB (BS x H)
  float* xproj = (float*)(ws + (104u << 20));           // 8 MB  (BS x E)
  float* hb0   = (float*)(ws + (112u << 20));           // 128 KB ping-pong h
  float* hb1   = (float*)(ws + (112u << 20) + (128u << 10));
  unsigned int* bars = (unsigned int*)(ws + (113u << 20));

  init_state<<<64, 256, 0, stream>>>(h00, h01, hb0, hb1, bars);
  embed_kernel<<<BS_, 256, 0, stream>>>(tokens, emb, x0);

  // xz0 = x0 @ W0 + b0
  gemm_wmma_f32<<<dim3(G4_ / 64, BS_ / 64), 256, 0, stream>>>(
      x0, W0, b0, xzb, BS_, G4_, E_, 0);
  lstm_seq<<<64, 128, 0, stream>>>(xzb, U0, c00, tokens, hb0, out0, bars + 0);

  // xz1 = out0 @ W1 + b1  (xz buffer reused; layer-0 recurrence already done)
  gemm_wmma_f32<<<dim3(G4_ / 64, BS_ / 64), 256, 0, stream>>>(
      out0, W1, b1, xzb, BS_, G4_, H_, 0);
  lstm_seq<<<64, 128, 0, stream>>>(xzb, U1, c01, tokens, hb1, out1, bars + 1);

  // xproj = out1 @ proj_w + proj_b
  gemm_wmma_f32<<<dim3(E_ / 64, BS_ / 64), 256, 0, stream>>>(
      out1, pw, pb, xproj, BS_, E_, H_, 0);

  // logits = xproj @ emb^T  (BT path)
  gemm_wmma_f32<<<dim3(V_ / 64, BS_ / 64), 256, 0, stream>>>(
      xproj, emb, nullptr, logits, BS_, V_, E_, 1);
}